// S2SProteinFeatures_48000554500378
// MI455X (gfx1250) — compile-verified
//
#include <hip/hip_runtime.h>
#include <math.h>

#define B_ 8
#define N_ 2048
#define K_ 30
#define NODE_F_ 128
#define EDGE_F_ 128
#define EDGE_IN_ 39
#define EIN_PAD 40

typedef float v2f __attribute__((ext_vector_type(2)));
typedef float v8f __attribute__((ext_vector_type(8)));

__device__ __forceinline__ float sgnf(float x) { return (x > 0.f) ? 1.f : ((x < 0.f) ? -1.f : 0.f); }
__device__ __forceinline__ float dot3(const float* a, const float* b) {
    return a[0]*b[0] + a[1]*b[1] + a[2]*b[2];
}
__device__ __forceinline__ void cross3(const float* a, const float* b, float* o) {
    o[0] = a[1]*b[2] - a[2]*b[1];
    o[1] = a[2]*b[0] - a[0]*b[2];
    o[2] = a[0]*b[1] - a[1]*b[0];
}
__device__ __forceinline__ void norm3(float* v) {
    float n = sqrtf(dot3(v, v));
    float inv = 1.f / fmaxf(n, 1e-12f);
    v[0] *= inv; v[1] *= inv; v[2] *= inv;
}

// ---------------------------------------------------------------------------
// Kernel 1: per-node orientation frames Of[b,n,9] = rows {o1, n2, o1 x n2}
// (padded: zero for n==0 and n>=N-2, matching jnp.pad((1,2)))
// ---------------------------------------------------------------------------
__global__ void of_kernel(const float* __restrict__ X, float* __restrict__ Of) {
    int idx = blockIdx.x * blockDim.x + threadIdx.x;
    if (idx >= B_ * N_) return;
    int n = idx % N_;
    float o[9] = {0.f,0.f,0.f,0.f,0.f,0.f,0.f,0.f,0.f};
    if (n >= 1 && n <= N_ - 3) {
        float P[3][3];
        #pragma unroll
        for (int q = 0; q < 3; q++) {
            int off = ((idx - 1 + q) * 4 + 1) * 3;   // Xca of node n-1+q
            P[q][0] = X[off]; P[q][1] = X[off + 1]; P[q][2] = X[off + 2];
        }
        float u2[3] = {P[1][0]-P[0][0], P[1][1]-P[0][1], P[1][2]-P[0][2]};
        float u1[3] = {P[2][0]-P[1][0], P[2][1]-P[1][1], P[2][2]-P[1][2]};
        norm3(u2); norm3(u1);
        float n2[3]; cross3(u2, u1, n2); norm3(n2);
        float o1[3] = {u2[0]-u1[0], u2[1]-u1[1], u2[2]-u1[2]};
        norm3(o1);
        float t3[3]; cross3(o1, n2, t3);
        o[0]=o1[0]; o[1]=o1[1]; o[2]=o1[2];
        o[3]=n2[0]; o[4]=n2[1]; o[5]=n2[2];
        o[6]=t3[0]; o[7]=t3[1]; o[8]=t3[2];
    }
    #pragma unroll
    for (int q = 0; q < 9; q++) Of[idx * 9 + q] = o[q];
}

// ---------------------------------------------------------------------------
// Kernel 2: masked pairwise distances + top-K smallest (== top_k of -D_adjust)
// One block of 256 threads per (b, i) row. Row cached in LDS; 30 rounds of
// packed u64 argmin (value bits || index) -> deterministic lowest-index ties.
// ---------------------------------------------------------------------------
__global__ void topk_kernel(const float* __restrict__ X, const float* __restrict__ mask,
                            float* __restrict__ Dn, int* __restrict__ Eix,
                            float* __restrict__ EixF) {
    __shared__ float sD[N_];
    __shared__ float smax[256];
    __shared__ unsigned long long sred[256];
    int tid = threadIdx.x;
    int idx = blockIdx.x;           // b*N + i
    int b = idx / N_, i = idx % N_;
    int offi = ((b * N_ + i) * 4 + 1) * 3;
    float xi0 = X[offi], xi1 = X[offi + 1], xi2 = X[offi + 2];
    float mi = mask[b * N_ + i];

    float lmax = 0.f;
    for (int j = tid; j < N_; j += 256) {
        int offj = ((b * N_ + j) * 4 + 1) * 3;
        float dx = X[offj] - xi0, dy = X[offj + 1] - xi1, dz = X[offj + 2] - xi2;
        float m2 = mi * mask[b * N_ + j];
        float Dv = m2 * sqrtf(dx*dx + dy*dy + dz*dz + 1e-6f);
        sD[j] = Dv;
        lmax = fmaxf(lmax, Dv);
    }
    smax[tid] = lmax;
    __syncthreads();
    for (int s = 128; s > 0; s >>= 1) {
        if (tid < s) smax[tid] = fmaxf(smax[tid], smax[tid + s]);
        __syncthreads();
    }
    float Dmax = smax[0];
    for (int j = tid; j < N_; j += 256) {
        float m2 = mi * mask[b * N_ + j];
        sD[j] += (1.f - m2) * Dmax;   // D_adjust
    }
    __syncthreads();

    for (int k = 0; k < K_; k++) {
        unsigned long long best = ~0ull;
        for (int j = tid; j < N_; j += 256) {
            unsigned long long key =
                ((unsigned long long)__float_as_uint(sD[j]) << 32) | (unsigned int)j;
            best = (key < best) ? key : best;
        }
        sred[tid] = best;
        __syncthreads();
        for (int s = 128; s > 0; s >>= 1) {
            if (tid < s) { if (sred[tid + s] < sred[tid]) sred[tid] = sred[tid + s]; }
            __syncthreads();
        }
        if (tid == 0) {
            unsigned long long w = sred[0];
            int jw = (int)(w & 0xffffffffull);
            float dv = __uint_as_float((unsigned int)(w >> 32));
            int o = (b * N_ + i) * K_ + k;
            Dn[o] = dv;                 // D_neighbors
            Eix[o] = jw;
            EixF[o] = (float)jw;        // E_idx output slice
            sD[jw] = INFINITY;          // remove from future rounds
        }
        __syncthreads();
    }
}

// ---------------------------------------------------------------------------
// Kernel 3: node features (dihedrals) -> 6x128 GEMV -> LayerNorm.
// One block (128 threads) per node; threads 0..2 compute the 3 dihedrals.
// ---------------------------------------------------------------------------
__global__ void node_kernel(const float* __restrict__ X,
                            const float* __restrict__ Wn, const float* __restrict__ bn,
                            const float* __restrict__ gn, const float* __restrict__ betan,
                            float* __restrict__ Vout) {
    int idx = blockIdx.x;            // b*N + n
    int n = idx % N_;
    int b = idx / N_;
    int tid = threadIdx.x;
    __shared__ float feat[6];
    __shared__ float red[128];

    if (tid < 3) {
        int m = 3 * n + tid;         // index into padded dihedral array (len 3N)
        float Dv = 0.f;
        if (m >= 1 && m <= 3 * N_ - 3) {
            int t = m - 1;
            float P[4][3];
            #pragma unroll
            for (int q = 0; q < 4; q++) {
                int qq = t + q;
                int resi = qq / 3, atom = qq % 3;
                int off = ((b * N_ + resi) * 4 + atom) * 3;
                P[q][0] = X[off]; P[q][1] = X[off + 1]; P[q][2] = X[off + 2];
            }
            float u2[3] = {P[1][0]-P[0][0], P[1][1]-P[0][1], P[1][2]-P[0][2]};
            float u1[3] = {P[2][0]-P[1][0], P[2][1]-P[1][1], P[2][2]-P[1][2]};
            float u0[3] = {P[3][0]-P[2][0], P[3][1]-P[2][1], P[3][2]-P[2][2]};
            norm3(u2); norm3(u1); norm3(u0);
            float n2[3], n1[3];
            cross3(u2, u1, n2); norm3(n2);
            cross3(u1, u0, n1); norm3(n1);
            float cosD = dot3(n2, n1);
            cosD = fminf(fmaxf(cosD, -1.f + 1e-7f), 1.f - 1e-7f);
            Dv = sgnf(dot3(u2, n1)) * acosf(cosD);
        }
        feat[tid]     = cosf(Dv);
        feat[tid + 3] = sinf(Dv);
    }
    __syncthreads();

    float x = bn[tid];
    #pragma unroll
    for (int c = 0; c < 6; c++) x += feat[c] * Wn[c * NODE_F_ + tid];

    red[tid] = x;
    __syncthreads();
    for (int s = 64; s > 0; s >>= 1) {
        if (tid < s) red[tid] += red[tid + s];
        __syncthreads();
    }
    float mu = red[0] * (1.f / 128.f);
    __syncthreads();
    float d = x - mu;
    red[tid] = d * d;
    __syncthreads();
    for (int s = 64; s > 0; s >>= 1) {
        if (tid < s) red[tid] += red[tid + s];
        __syncthreads();
    }
    float var = red[0] * (1.f / 128.f);
    Vout[idx * NODE_F_ + tid] = d * rsqrtf(var + 1e-5f) * gn[tid] + betan[tid];
}

// ---------------------------------------------------------------------------
// Kernel 4: edge features -> (16x40)x(40x128) GEMM via V_WMMA_F32_16X16X4_F32
// -> LayerNorm -> store. One wave per 16-edge group, 8 waves per block.
// LDS: W_edge padded 40x128 (20KB, block-shared) + per-wave A tiles (20KB).
// ---------------------------------------------------------------------------
__global__ void __launch_bounds__(256)
edge_kernel(const float* __restrict__ X,
            const float* __restrict__ Of,
            const float* __restrict__ Dn,
            const int* __restrict__ Eix,
            const float* __restrict__ We,
            const float* __restrict__ be,
            const float* __restrict__ ge,
            const float* __restrict__ betae,
            float* __restrict__ Eout) {
    __shared__ float sW[EIN_PAD * EDGE_F_];     // K-major: sW[k*128 + f]
    __shared__ float sA[8][16 * EIN_PAD];       // per wave: 16 edges x 40 feats
    int tid = threadIdx.x;

    for (int x = tid; x < EIN_PAD * EDGE_F_; x += 256) {
        int r = x / EDGE_F_;
        sW[x] = (r < EDGE_IN_) ? We[x] : 0.f;   // We is (39,128) row-major
    }

    int wave = tid >> 5;
    int lane = tid & 31;
    int base_e = (blockIdx.x * 8 + wave) * 16;

    if (lane < 16) {
        int e = base_e + lane;
        int b = e / (N_ * K_);
        int rem = e % (N_ * K_);
        int i = rem / K_;
        int j = Eix[e];
        float Dnb = Dn[e];
        float* f = &sA[wave][lane * EIN_PAD];

        // positional encodings: freq_q = 10000^(-q/8)
        float d = (float)(j - i);
        #pragma unroll
        for (int q = 0; q < 8; q++) {
            float ang = d * expf(-1.15129254649702284f * (float)q);
            f[q]     = cosf(ang);
            f[8 + q] = sinf(ang);
        }
        // RBF: mu = linspace(0,20,16), sigma = 1.25
        #pragma unroll
        for (int m = 0; m < 16; m++) {
            float mu = (20.f / 15.f) * (float)m;
            float z = (Dnb - mu) * (1.f / 1.25f);
            f[16 + m] = expf(-z * z);
        }
        // orientation features
        float Om[9], On[9];
        #pragma unroll
        for (int q = 0; q < 9; q++) {
            Om[q] = Of[(b * N_ + i) * 9 + q];
            On[q] = Of[(b * N_ + j) * 9 + q];
        }
        int offi = ((b * N_ + i) * 4 + 1) * 3;
        int offj = ((b * N_ + j) * 4 + 1) * 3;
        float dX[3] = {X[offj] - X[offi], X[offj+1] - X[offi+1], X[offj+2] - X[offi+2]};
        float v[3];
        v[0] = Om[0]*dX[0] + Om[1]*dX[1] + Om[2]*dX[2];
        v[1] = Om[3]*dX[0] + Om[4]*dX[1] + Om[5]*dX[2];
        v[2] = Om[6]*dX[0] + Om[7]*dX[1] + Om[8]*dX[2];
        float nv = sqrtf(v[0]*v[0] + v[1]*v[1] + v[2]*v[2]);
        float inv = 1.f / fmaxf(nv, 1e-12f);
        f[32] = v[0]*inv; f[33] = v[1]*inv; f[34] = v[2]*inv;

        float R[9];                    // R = Om^T * On
        #pragma unroll
        for (int a = 0; a < 3; a++)
            #pragma unroll
            for (int l = 0; l < 3; l++)
                R[a*3 + l] = Om[0*3 + a]*On[0*3 + l] + Om[1*3 + a]*On[1*3 + l] + Om[2*3 + a]*On[2*3 + l];
        float Rxx = R[0], Ryy = R[4], Rzz = R[8];
        float mx = 0.5f * sqrtf(fabsf(1.f + Rxx - Ryy - Rzz));
        float my = 0.5f * sqrtf(fabsf(1.f - Rxx + Ryy - Rzz));
        float mz = 0.5f * sqrtf(fabsf(1.f - Rxx - Ryy + Rzz));
        float qx = sgnf(R[7] - R[5]) * mx;
        float qy = sgnf(R[2] - R[6]) * my;
        float qz = sgnf(R[3] - R[1]) * mz;
        float qw = 0.5f * sqrtf(fmaxf(1.f + Rxx + Ryy + Rzz, 0.f));
        float qn = sqrtf(qx*qx + qy*qy + qz*qz + qw*qw);
        float qi = 1.f / fmaxf(qn, 1e-12f);
        f[35] = qx*qi; f[36] = qy*qi; f[37] = qz*qi; f[38] = qw*qi;
        f[39] = 0.f;                   // padding feature
    }
    __syncthreads();

    // --- WMMA f32 16x16x4: A = 16 edges x 40 feats, B = 40 x 128 weights ---
    int half = lane >> 4;              // A: lanes 0-15 hold K=0/1, 16-31 K=2/3
    int col  = lane & 15;              // A row M / B,C,D column N
    const float* A = sA[wave];

    v2f av[10];
    #pragma unroll
    for (int s = 0; s < 10; s++) {
        av[s].x = A[col * EIN_PAD + 4*s     + 2*half];
        av[s].y = A[col * EIN_PAD + 4*s + 1 + 2*half];
    }

    v8f acc[8];
    float s1[8] = {0.f,0.f,0.f,0.f,0.f,0.f,0.f,0.f};
    float s2[8] = {0.f,0.f,0.f,0.f,0.f,0.f,0.f,0.f};
    #pragma unroll
    for (int t = 0; t < 8; t++) {
        int fcol = t * 16 + col;
        float bias = be[fcol];
        v8f c;
        #pragma unroll
        for (int r = 0; r < 8; r++) c[r] = bias;
        #pragma unroll
        for (int s = 0; s < 10; s++) {
            v2f bv;
            bv.x = sW[(4*s     + 2*half) * EDGE_F_ + fcol];
            bv.y = sW[(4*s + 1 + 2*half) * EDGE_F_ + fcol];
            c = __builtin_amdgcn_wmma_f32_16x16x4_f32(
                    false, av[s], false, bv, (short)0, c, false, false);
        }
        acc[t] = c;
        #pragma unroll
        for (int r = 0; r < 8; r++) { s1[r] += c[r]; s2[r] += c[r]*c[r]; }
    }

    // reduce over the 16 lanes of each half (row M = r + 8*half holds 128 feats)
    #pragma unroll
    for (int r = 0; r < 8; r++) {
        #pragma unroll
        for (int m = 1; m < 16; m <<= 1) {
            s1[r] += __shfl_xor(s1[r], m, 32);
            s2[r] += __shfl_xor(s2[r], m, 32);
        }
    }
    float mu[8], rs[8];
    #pragma unroll
    for (int r = 0; r < 8; r++) {
        mu[r] = s1[r] * (1.f / 128.f);
        float var = s2[r] * (1.f / 128.f) - mu[r]*mu[r];
        rs[r] = rsqrtf(var + 1e-5f);
    }
    #pragma unroll
    for (int t = 0; t < 8; t++) {
        int fcol = t * 16 + col;
        float gv = ge[fcol], bv2 = betae[fcol];
        #pragma unroll
        for (int r = 0; r < 8; r++) {
            int row = base_e + r + 8*half;     // edge index
            Eout[row * EDGE_F_ + fcol] = (acc[t][r] - mu[r]) * rs[r] * gv + bv2;
        }
    }
}

// ---------------------------------------------------------------------------
extern "C" void kernel_launch(void* const* d_in, const int* in_sizes, int n_in,
                              void* d_out, int out_size, void* d_ws, size_t ws_size,
                              hipStream_t stream) {
    const float* X     = (const float*)d_in[0];
    const float* mask  = (const float*)d_in[1];
    const float* Wn    = (const float*)d_in[2];
    const float* bn    = (const float*)d_in[3];
    const float* We    = (const float*)d_in[4];
    const float* be    = (const float*)d_in[5];
    const float* gn    = (const float*)d_in[6];
    const float* betan = (const float*)d_in[7];
    const float* ge    = (const float*)d_in[8];
    const float* betae = (const float*)d_in[9];

    // outputs: V (B,N,128) | E (B,N,K,128) | E_idx (B,N,K)
    float* Vout = (float*)d_out;
    float* Eout = Vout + (size_t)B_ * N_ * NODE_F_;
    float* EixF = Eout + (size_t)B_ * N_ * K_ * EDGE_F_;

    // workspace: Of (B,N,9) | D_neighbors (B,N,K) | E_idx int (B,N,K)
    float* Of  = (float*)d_ws;
    float* Dnb = Of + (size_t)B_ * N_ * 9;
    int*   Eix = (int*)(Dnb + (size_t)B_ * N_ * K_);

    of_kernel<<<(B_ * N_ + 255) / 256, 256, 0, stream>>>(X, Of);
    topk_kernel<<<B_ * N_, 256, 0, stream>>>(X, mask, Dnb, Eix, EixF);
    node_kernel<<<B_ * N_, 128, 0, stream>>>(X, Wn, bn, gn, betan, Vout);
    edge_kernel<<<(B_ * N_ * K_) / (16 * 8), 256, 0, stream>>>(
        X, Of, Dnb, Eix, We, be, ge, betae, Eout);
}